// SimpleGatedGNN_70540542869535
// MI455X (gfx1250) — compile-verified
//
#include <hip/hip_runtime.h>

// ---------------------------------------------------------------------------
// SimpleGatedGNN on MI455X (gfx1250): f32 WMMA (16x16x4) GEMMs, fused GRU,
// atomic scatter-add for the graph aggregation. Wave32 layouts per CDNA5 ISA.
// ---------------------------------------------------------------------------

typedef __attribute__((ext_vector_type(2))) float v2f;
typedef __attribute__((ext_vector_type(8))) float v8f;

#define N_NODES 100000
#define N_EDGES 600000
#define DIM     128
#define DOUT    384

__device__ __forceinline__ float sigmoidf_(float x) {
    return 1.0f / (1.0f + __expf(-x));
}

// ---------------------------------------------------------------------------
// Generic GEMM: C[nrows, NCOLS] = act_out( act_in(A[nrows,128]) @ W[NCOLS,128]^T + bias )
// One block = 256 threads = 8 waves, owns 16 rows. Each wave computes
// NCOLS/128 column tiles of 16x16 via V_WMMA_F32_16X16X4_F32, K-loop of 32.
// ---------------------------------------------------------------------------
template <int NCOLS, int ACT_IN, int ACT_OUT>
__global__ __launch_bounds__(256) void gemm_wmma_k128(
    const float* __restrict__ A, const float* __restrict__ W,
    const float* __restrict__ bias, float* __restrict__ C)
{
    __shared__ float As[16 * DIM];  // 8 KB, contiguous 16-row tile of A

    const size_t tileBase = (size_t)blockIdx.x * (16 * DIM);
    for (int i = threadIdx.x; i < 16 * DIM; i += 256) {
        float v = A[tileBase + i];
        As[i] = ACT_IN ? fmaxf(v, 0.0f) : v;
    }
    __syncthreads();

    const int lane  = threadIdx.x & 31;
    const int wave  = threadIdx.x >> 5;
    const int mrow  = lane & 15;            // A-frag row (M)
    const int kh    = (lane >> 4) << 1;     // K sub-offset: 0 (lanes 0-15) / 2 (16-31)
    const int half8 = (lane >> 4) << 3;     // C-frag row offset

    constexpr int TPW = NCOLS / 128;        // column tiles per wave
    #pragma unroll
    for (int t = 0; t < TPW; ++t) {
        const int n = (wave + t * 8) * 16 + (lane & 15);   // global output column
        const float* __restrict__ Wn = W + (size_t)n * DIM;

        v8f acc = {0.f, 0.f, 0.f, 0.f, 0.f, 0.f, 0.f, 0.f};
        #pragma unroll 8
        for (int k = 0; k < DIM; k += 4) {
            v2f a = *(const v2f*)(As + mrow * DIM + k + kh);  // ds_load_b64
            v2f b = *(const v2f*)(Wn + k + kh);               // global_load_b64 (L2-hot)
            acc = __builtin_amdgcn_wmma_f32_16x16x4_f32(
                false, a, false, b, (short)0, acc, false, false);
        }

        const float bn = bias ? bias[n] : 0.0f;
        #pragma unroll
        for (int v = 0; v < 8; ++v) {
            size_t gr  = (size_t)blockIdx.x * 16 + v + half8;
            float  val = acc[v] + bn;
            if (ACT_OUT) val = fmaxf(val, 0.0f);
            C[gr * NCOLS + n] = val;
        }
    }
}

// ---------------------------------------------------------------------------
// Fused GRU cell: gi = agg @ Wih^T + bih ; gh = h @ Whh^T + bhh (both 128->384),
// gate math, in-place h update. Block owns 16 rows; both gate tiles in LDS.
// ---------------------------------------------------------------------------
__global__ __launch_bounds__(256) void gru_wmma(
    const float* __restrict__ agg, float* __restrict__ h,
    const float* __restrict__ Wih, const float* __restrict__ Whh,
    const float* __restrict__ bih, const float* __restrict__ bhh)
{
    __shared__ float smem[2 * 16 * DIM + 2 * 16 * DOUT];  // 64 KB
    float* Aag = smem;                 // 16 x 128
    float* Ah  = smem + 2048;          // 16 x 128
    float* Gi  = smem + 4096;          // 16 x 384
    float* Gh  = smem + 4096 + 6144;   // 16 x 384

    const size_t tileBase = (size_t)blockIdx.x * 2048;
    for (int i = threadIdx.x; i < 2048; i += 256) {
        Aag[i] = agg[tileBase + i];
        Ah[i]  = h[tileBase + i];
    }
    __syncthreads();

    const int lane  = threadIdx.x & 31;
    const int wave  = threadIdx.x >> 5;
    const int mrow  = lane & 15;
    const int kh    = (lane >> 4) << 1;
    const int half8 = (lane >> 4) << 3;

    #pragma unroll
    for (int t = 0; t < 3; ++t) {              // 24 column tiles over 8 waves
        const int n = (wave + t * 8) * 16 + (lane & 15);
        const float* __restrict__ Win = Wih + (size_t)n * DIM;
        const float* __restrict__ Whn = Whh + (size_t)n * DIM;

        v8f acc_i = {0.f, 0.f, 0.f, 0.f, 0.f, 0.f, 0.f, 0.f};
        v8f acc_h = {0.f, 0.f, 0.f, 0.f, 0.f, 0.f, 0.f, 0.f};
        #pragma unroll 8
        for (int k = 0; k < DIM; k += 4) {
            v2f aA = *(const v2f*)(Aag + mrow * DIM + k + kh);
            v2f aH = *(const v2f*)(Ah  + mrow * DIM + k + kh);
            v2f bi = *(const v2f*)(Win + k + kh);
            v2f bh = *(const v2f*)(Whn + k + kh);
            acc_i = __builtin_amdgcn_wmma_f32_16x16x4_f32(
                false, aA, false, bi, (short)0, acc_i, false, false);
            acc_h = __builtin_amdgcn_wmma_f32_16x16x4_f32(
                false, aH, false, bh, (short)0, acc_h, false, false);
        }

        const float bin = bih[n];
        const float bhn = bhh[n];
        #pragma unroll
        for (int v = 0; v < 8; ++v) {
            const int row = v + half8;
            Gi[row * DOUT + n] = acc_i[v] + bin;
            Gh[row * DOUT + n] = acc_h[v] + bhn;
        }
    }
    __syncthreads();

    // Gate math + in-place h update (torch GRUCell gate order r,z,n)
    for (int i = threadIdx.x; i < 2048; i += 256) {
        const int row = i >> 7;
        const int c   = i & 127;
        const float ir  = Gi[row * DOUT + c];
        const float iz  = Gi[row * DOUT + c + 128];
        const float inn = Gi[row * DOUT + c + 256];
        const float hr  = Gh[row * DOUT + c];
        const float hz  = Gh[row * DOUT + c + 128];
        const float hn  = Gh[row * DOUT + c + 256];
        const float hv  = Ah[row * DIM + c];
        const float r = sigmoidf_(ir + hr);
        const float z = sigmoidf_(iz + hz);
        const float nn = tanhf(inn + r * hn);
        h[tileBase + i] = (1.0f - z) * nn + z * hv;
    }
}

// ---------------------------------------------------------------------------
// Scatter-add: one wave per edge; each lane moves 16 B with float atomics.
// ---------------------------------------------------------------------------
__global__ __launch_bounds__(256) void scatter_add(
    const float* __restrict__ msg, const long long* __restrict__ src,
    const long long* __restrict__ dst, float* __restrict__ agg)
{
    const int e = blockIdx.x * 8 + (threadIdx.x >> 5);
    if (e >= N_EDGES) return;
    const int lane = threadIdx.x & 31;
    const long long s = src[e];
    const long long d = dst[e];
    const float4 v = ((const float4*)(msg + (size_t)s * DIM))[lane];
    float* ad = agg + (size_t)d * DIM + lane * 4;
    atomicAdd(ad + 0, v.x);
    atomicAdd(ad + 1, v.y);
    atomicAdd(ad + 2, v.z);
    atomicAdd(ad + 3, v.w);
}

__global__ __launch_bounds__(256) void zero_f4(float4* __restrict__ p, int n4)
{
    const int i = blockIdx.x * blockDim.x + threadIdx.x;
    if (i < n4) p[i] = make_float4(0.f, 0.f, 0.f, 0.f);
}

// ---------------------------------------------------------------------------
extern "C" void kernel_launch(void* const* d_in, const int* in_sizes, int n_in,
                              void* d_out, int out_size, void* d_ws, size_t ws_size,
                              hipStream_t stream)
{
    const float*     x     = (const float*)d_in[0];
    const long long* edges = (const long long*)d_in[1];  // int64 [2, N_EDGES]
    const float*     W1    = (const float*)d_in[2];
    const float*     b1    = (const float*)d_in[3];
    const float*     Wconv = (const float*)d_in[4];      // [3,128,128]
    const float*     Wih   = (const float*)d_in[5];
    const float*     Whh   = (const float*)d_in[6];
    const float*     bih   = (const float*)d_in[7];
    const float*     bhh   = (const float*)d_in[8];
    const float*     W2    = (const float*)d_in[9];
    const float*     b2    = (const float*)d_in[10];

    const long long* src = edges;
    const long long* dst = edges + N_EDGES;

    float* h   = (float*)d_ws;
    float* msg = h   + (size_t)N_NODES * DIM;
    float* agg = msg + (size_t)N_NODES * DIM;

    const int rowBlocks = N_NODES / 16;  // 6250 (exact)

    // linear1 + relu
    gemm_wmma_k128<128, 0, 1><<<rowBlocks, 256, 0, stream>>>(x, W1, b1, h);

    for (int l = 0; l < 3; ++l) {
        // message transform: m = h @ Wconv[l]^T
        gemm_wmma_k128<128, 0, 0><<<rowBlocks, 256, 0, stream>>>(
            h, Wconv + (size_t)l * DIM * DIM, nullptr, msg);
        // agg = segment_sum(m[src], dst)
        zero_f4<<<(N_NODES * DIM / 4 + 255) / 256, 256, 0, stream>>>(
            (float4*)agg, N_NODES * DIM / 4);
        scatter_add<<<(N_EDGES + 7) / 8, 256, 0, stream>>>(msg, src, dst, agg);
        // GRU cell (in-place h update)
        gru_wmma<<<rowBlocks, 256, 0, stream>>>(agg, h, Wih, Whh, bih, bhh);
    }

    // linear2 on relu(h)
    gemm_wmma_k128<384, 1, 0><<<rowBlocks, 256, 0, stream>>>(
        h, W2, b2, (float*)d_out);
}